// uvNet_28449863369525
// MI455X (gfx1250) — compile-verified
//
#include <hip/hip_runtime.h>
#include <math.h>

constexpr int CN = 128;          // channels
constexpr int HN = 128, WN = 128;
constexpr int PN = HN * WN;      // 16384 pixels
constexpr int KKN = 9;           // 3x3 taps
constexpr int RD = CN * KKN;     // 1152 reduction dim for DCN GEMM
constexpr int NBATCH = 2;
constexpr float ATT_SCALE = 0.08838834764831845f;  // 1/sqrt(128)

constexpr int KCH = 16;          // K chunk per LDS stage
constexpr int SA = 20;           // A LDS row stride (pad; 80B = 16B-aligned rows)
constexpr int SB = 136;          // B LDS row stride (pad vs 128; 544B = 16B-aligned)
constexpr int NCHUNK = 256;      // split-K chunk (n) per WG in attn logits

typedef __attribute__((ext_vector_type(2))) float v2f;
typedef __attribute__((ext_vector_type(8))) float v8f;
typedef int v4i_vs __attribute__((vector_size(16)));   // matches builtin param type

// CDNA5 async global->LDS path (ASYNCcnt-tracked), if this toolchain declares it.
#if __has_builtin(__builtin_amdgcn_global_load_async_to_lds_b128) && \
    __has_builtin(__builtin_amdgcn_global_load_async_to_lds_b32)  && \
    __has_builtin(__builtin_amdgcn_s_wait_asynccnt)
#define USE_ASYNC_LDS 1
#define GP128(p) ((v4i_vs*)(p))
#define LP128(p) ((v4i_vs*)(p))
#define GP32(p)  ((int*)(p))
#define LP32(p)  ((int*)(p))
#else
#define USE_ASYNC_LDS 0
#endif

// ---------------------------------------------------------------------------
// Offset conv: standard 3x3 conv, 18 output channels, zero padding.
// grid = (PN/256, 18, 4) ; z = t*2+b (t: 0=u,1=v)
// ---------------------------------------------------------------------------
__global__ __launch_bounds__(256)
void offset_conv_kernel(const float* __restrict__ u, const float* __restrict__ v,
                        const float* __restrict__ offw, const float* __restrict__ offb,
                        float* __restrict__ off_all) {
  int tid = threadIdx.x;
  int p   = blockIdx.x * 256 + tid;
  int ch  = blockIdx.y;                 // 0..17
  int img = blockIdx.z;                 // 0..3
  int t = img >> 1, bb = img & 1;
  const float* x = (t ? v : u) + (size_t)bb * CN * PN;
  int y = p >> 7, xx = p & (WN - 1);
  float acc = offb[ch];
  for (int cc = 0; cc < CN; ++cc) {
    const float* xc = x + (size_t)cc * PN;
    const float* wp = offw + ((size_t)ch * CN + cc) * 9;
    #pragma unroll
    for (int ki = 0; ki < 3; ++ki) {
      int yy = y + ki - 1;
      if (yy < 0 || yy >= HN) continue;
      #pragma unroll
      for (int kj = 0; kj < 3; ++kj) {
        int x2 = xx + kj - 1;
        if (x2 < 0 || x2 >= WN) continue;
        acc = fmaf(xc[yy * WN + x2], wp[ki * 3 + kj], acc);
      }
    }
  }
  off_all[((size_t)img * 18 + ch) * PN + p] = acc;
}

// ---------------------------------------------------------------------------
// Deformable im2col: S[cc*9+k][p] = bilinear sample of x[cc] at offset tap k.
// grid = (PN/256, 9, 128)
// ---------------------------------------------------------------------------
__global__ __launch_bounds__(256)
void build_S_kernel(const float* __restrict__ x,    // [CN][PN] one image
                    const float* __restrict__ off,  // [18][PN] this image
                    float* __restrict__ S) {        // [RD][PN]
  int tid = threadIdx.x;
  int p  = blockIdx.x * 256 + tid;
  int k  = blockIdx.y;                  // 0..8
  int cc = blockIdx.z;                  // 0..127
  int y = p >> 7, xx = p & (WN - 1);
  float dy = off[(2 * k + 0) * PN + p];
  float dx = off[(2 * k + 1) * PN + p];
  float py = (float)y  + (float)(k / 3) - 1.0f + dy;
  float px = (float)xx + (float)(k % 3) - 1.0f + dx;
  float y0f = floorf(py), x0f = floorf(px);
  float wy = py - y0f, wx = px - x0f;
  int y0 = (int)y0f, x0 = (int)x0f;
  const float* xc = x + (size_t)cc * PN;
  auto g = [&](int yi, int xi) -> float {
    if (yi < 0 || yi >= HN || xi < 0 || xi >= WN) return 0.0f;
    return xc[yi * WN + xi];
  };
  float v00 = g(y0, x0),     v01 = g(y0, x0 + 1);
  float v10 = g(y0 + 1, x0), v11 = g(y0 + 1, x0 + 1);
  float val = v00 * (1.0f - wy) * (1.0f - wx) + v01 * (1.0f - wy) * wx +
              v10 * wy * (1.0f - wx)          + v11 * wy * wx;
  S[((size_t)cc * KKN + k) * PN + p] = val;
}

// ---------------------------------------------------------------------------
// GEMM: Cm[128][16384] = A[128][K] * Bm[K][16384] (+bias), f32 WMMA 16x16x4.
// 256 threads = 8 waves; WG tile 128x128; each wave a 128x16 stripe.
// Async path: double-buffered GLOBAL_LOAD_ASYNC_TO_LDS (ASYNCcnt pipelined).
// grid = (PN/128)
// ---------------------------------------------------------------------------
__global__ __launch_bounds__(256)
void gemm128_wmma_kernel(const float* __restrict__ A, const float* __restrict__ Bm,
                         const float* __restrict__ bias, float* __restrict__ Cm,
                         int K) {
  int tid  = threadIdx.x;
  int lane = tid & 31;
  int wv   = tid >> 5;
  int n0   = blockIdx.x * 128;
  int lcol = lane & 15;
  int half = lane >> 4;
  int half2 = half * 2;
  int ncol = n0 + wv * 16 + lcol;

  v8f acc[8] = {};

  int la_m = tid >> 1, la_k = (tid & 1) * 8;     // A loader: 2 thr/row x 8 floats
  int lb_r = tid >> 4, lb_j = (tid & 15) * 8;    // B loader: 16 thr/row x 8 floats

#if USE_ASYNC_LDS
  __shared__ __align__(16) float Alds[2][128 * SA];
  __shared__ __align__(16) float Blds[2][KCH * SB];
  const float* agp = A + (size_t)la_m * K + la_k;
  const float* bgp = Bm + (size_t)lb_r * PN + n0 + lb_j;
  float* al = &Alds[0][la_m * SA + la_k];
  float* bl = &Blds[0][lb_r * SB + lb_j];
  auto issue = [&](int k0, int buf) {
    float* a = al + buf * (128 * SA);
    float* b = bl + buf * (KCH * SB);
    __builtin_amdgcn_global_load_async_to_lds_b128(GP128(agp + k0),     LP128(a),     0, 0);
    __builtin_amdgcn_global_load_async_to_lds_b128(GP128(agp + k0 + 4), LP128(a + 4), 0, 0);
    __builtin_amdgcn_global_load_async_to_lds_b128(GP128(bgp + (size_t)k0 * PN),     LP128(b),     0, 0);
    __builtin_amdgcn_global_load_async_to_lds_b128(GP128(bgp + (size_t)k0 * PN + 4), LP128(b + 4), 0, 0);
  };
  issue(0, 0);
  for (int k0 = 0; k0 < K; k0 += KCH) {
    int buf = (k0 / KCH) & 1;
    if (k0 + KCH < K) {
      issue(k0 + KCH, buf ^ 1);
      __builtin_amdgcn_s_wait_asynccnt(4);   // chunk k0 landed; next in flight
    } else {
      __builtin_amdgcn_s_wait_asynccnt(0);
    }
    __syncthreads();
    const float* Ab = &Alds[buf][0];
    const float* Bb = &Blds[buf][0];
    #pragma unroll
    for (int kk = 0; kk < KCH; kk += 4) {
      v2f bfrag;
      bfrag.x = Bb[(kk + half2 + 0) * SB + wv * 16 + lcol];
      bfrag.y = Bb[(kk + half2 + 1) * SB + wv * 16 + lcol];
      #pragma unroll
      for (int mt = 0; mt < 8; ++mt) {
        v2f afrag = *(const v2f*)&Ab[(mt * 16 + lcol) * SA + kk + half2];
        acc[mt] = __builtin_amdgcn_wmma_f32_16x16x4_f32(
            false, afrag, false, bfrag, (short)0, acc[mt], false, false);
      }
    }
    __syncthreads();   // all waves done reading buf before it is refilled
  }
#else
  __shared__ __align__(16) float Alds[128 * SA];
  __shared__ __align__(16) float Blds[KCH * SB];
  for (int k0 = 0; k0 < K; k0 += KCH) {
    __syncthreads();
    const float* ap = A + (size_t)la_m * K + k0 + la_k;
    float4 a0 = *(const float4*)(ap);
    float4 a1 = *(const float4*)(ap + 4);
    float* ad = &Alds[la_m * SA + la_k];
    ad[0] = a0.x; ad[1] = a0.y; ad[2] = a0.z; ad[3] = a0.w;
    ad[4] = a1.x; ad[5] = a1.y; ad[6] = a1.z; ad[7] = a1.w;
    const float* bp = Bm + (size_t)(k0 + lb_r) * PN + n0 + lb_j;
    float4 b0 = *(const float4*)(bp);
    float4 b1 = *(const float4*)(bp + 4);
    float* bd = &Blds[lb_r * SB + lb_j];
    bd[0] = b0.x; bd[1] = b0.y; bd[2] = b0.z; bd[3] = b0.w;
    bd[4] = b1.x; bd[5] = b1.y; bd[6] = b1.z; bd[7] = b1.w;
    if (k0 + KCH < K) {
      __builtin_prefetch(bp + (size_t)KCH * PN, 0, 1);   // global_prefetch_b8
      __builtin_prefetch(ap + KCH, 0, 1);
    }
    __syncthreads();
    #pragma unroll
    for (int kk = 0; kk < KCH; kk += 4) {
      v2f bfrag;
      bfrag.x = Blds[(kk + half2 + 0) * SB + wv * 16 + lcol];
      bfrag.y = Blds[(kk + half2 + 1) * SB + wv * 16 + lcol];
      #pragma unroll
      for (int mt = 0; mt < 8; ++mt) {
        v2f afrag = *(const v2f*)&Alds[(mt * 16 + lcol) * SA + kk + half2];
        acc[mt] = __builtin_amdgcn_wmma_f32_16x16x4_f32(
            false, afrag, false, bfrag, (short)0, acc[mt], false, false);
      }
    }
  }
#endif

  #pragma unroll
  for (int mt = 0; mt < 8; ++mt) {
    int mbase = mt * 16 + half * 8;   // D: VGPR r -> row mbase+r, lane -> col
    #pragma unroll
    for (int r = 0; r < 8; ++r) {
      float bv = bias ? bias[mbase + r] : 0.0f;
      Cm[(size_t)(mbase + r) * PN + ncol] = acc[mt][r] + bv;
    }
  }
}

// ---------------------------------------------------------------------------
// Attention logits: attn[b][c0][c1] += scale * sum_n qu[b][c0][n]*kv[b][c1][n]
// Split-K over n with atomic accumulation. grid = (PN/NCHUNK, NBATCH)
// ---------------------------------------------------------------------------
__global__ __launch_bounds__(256)
void attn_logits_kernel(const float* __restrict__ qu, const float* __restrict__ kv,
                        float* __restrict__ attn) {
  int tid  = threadIdx.x;
  int lane = tid & 31;
  int wv   = tid >> 5;
  int lcol = lane & 15;
  int half = lane >> 4;
  int half2 = half * 2;
  int bb = blockIdx.y;
  const float* Aq = qu + (size_t)bb * CN * PN;
  const float* Bk = kv + (size_t)bb * CN * PN;
  float* ao = attn + (size_t)bb * CN * CN;
  int kstart = blockIdx.x * NCHUNK;

  v8f acc[8] = {};

  int la_m = tid >> 1, la_k = (tid & 1) * 8;
  int lb_j = tid >> 1, lb_r = (tid & 1) * 8;   // load kv[j][k..k+7], transpose to LDS

#if USE_ASYNC_LDS
  __shared__ __align__(16) float Alds[2][128 * SA];
  __shared__ __align__(16) float Blds[2][KCH * SB];
  const float* agp = Aq + (size_t)la_m * PN + la_k;
  const float* bgp = Bk + (size_t)lb_j * PN + lb_r;
  float* al = &Alds[0][la_m * SA + la_k];
  auto issue = [&](int k0, int buf) {
    float* a = al + buf * (128 * SA);
    __builtin_amdgcn_global_load_async_to_lds_b128(GP128(agp + k0),     LP128(a),     0, 0);
    __builtin_amdgcn_global_load_async_to_lds_b128(GP128(agp + k0 + 4), LP128(a + 4), 0, 0);
    #pragma unroll
    for (int i = 0; i < 8; ++i) {   // transposing scatter: one dword per copy
      __builtin_amdgcn_global_load_async_to_lds_b32(
          GP32(bgp + k0 + i), LP32(&Blds[buf][(lb_r + i) * SB + lb_j]), 0, 0);
    }
  };
  issue(kstart, 0);
  for (int k0 = kstart; k0 < kstart + NCHUNK; k0 += KCH) {
    int buf = ((k0 - kstart) / KCH) & 1;
    if (k0 + KCH < kstart + NCHUNK) {
      issue(k0 + KCH, buf ^ 1);
      __builtin_amdgcn_s_wait_asynccnt(10);
    } else {
      __builtin_amdgcn_s_wait_asynccnt(0);
    }
    __syncthreads();
    const float* Ab = &Alds[buf][0];
    const float* Bb = &Blds[buf][0];
    #pragma unroll
    for (int kk = 0; kk < KCH; kk += 4) {
      v2f bfrag;
      bfrag.x = Bb[(kk + half2 + 0) * SB + wv * 16 + lcol];
      bfrag.y = Bb[(kk + half2 + 1) * SB + wv * 16 + lcol];
      #pragma unroll
      for (int mt = 0; mt < 8; ++mt) {
        v2f afrag = *(const v2f*)&Ab[(mt * 16 + lcol) * SA + kk + half2];
        acc[mt] = __builtin_amdgcn_wmma_f32_16x16x4_f32(
            false, afrag, false, bfrag, (short)0, acc[mt], false, false);
      }
    }
    __syncthreads();
  }
#else
  __shared__ __align__(16) float Alds[128 * SA];
  __shared__ __align__(16) float Blds[KCH * SB];
  for (int k0 = kstart; k0 < kstart + NCHUNK; k0 += KCH) {
    __syncthreads();
    const float* ap = Aq + (size_t)la_m * PN + k0 + la_k;
    float4 a0 = *(const float4*)(ap);
    float4 a1 = *(const float4*)(ap + 4);
    float* ad = &Alds[la_m * SA + la_k];
    ad[0] = a0.x; ad[1] = a0.y; ad[2] = a0.z; ad[3] = a0.w;
    ad[4] = a1.x; ad[5] = a1.y; ad[6] = a1.z; ad[7] = a1.w;
    const float* bp = Bk + (size_t)lb_j * PN + k0 + lb_r;
    float4 b0 = *(const float4*)(bp);
    float4 b1 = *(const float4*)(bp + 4);
    Blds[(lb_r + 0) * SB + lb_j] = b0.x;
    Blds[(lb_r + 1) * SB + lb_j] = b0.y;
    Blds[(lb_r + 2) * SB + lb_j] = b0.z;
    Blds[(lb_r + 3) * SB + lb_j] = b0.w;
    Blds[(lb_r + 4) * SB + lb_j] = b1.x;
    Blds[(lb_r + 5) * SB + lb_j] = b1.y;
    Blds[(lb_r + 6) * SB + lb_j] = b1.z;
    Blds[(lb_r + 7) * SB + lb_j] = b1.w;
    __syncthreads();
    #pragma unroll
    for (int kk = 0; kk < KCH; kk += 4) {
      v2f bfrag;
      bfrag.x = Blds[(kk + half2 + 0) * SB + wv * 16 + lcol];
      bfrag.y = Blds[(kk + half2 + 1) * SB + wv * 16 + lcol];
      #pragma unroll
      for (int mt = 0; mt < 8; ++mt) {
        v2f afrag = *(const v2f*)&Alds[(mt * 16 + lcol) * SA + kk + half2];
        acc[mt] = __builtin_amdgcn_wmma_f32_16x16x4_f32(
            false, afrag, false, bfrag, (short)0, acc[mt], false, false);
      }
    }
  }
#endif

  int ncol = wv * 16 + lcol;
  #pragma unroll
  for (int mt = 0; mt < 8; ++mt) {
    int mbase = mt * 16 + half * 8;
    #pragma unroll
    for (int r = 0; r < 8; ++r) {
      atomicAdd(&ao[(size_t)(mbase + r) * CN + ncol], ATT_SCALE * acc[mt][r]);
    }
  }
}

// ---------------------------------------------------------------------------
// Row softmax over attn[.][128]. grid = NBATCH*CN blocks of 128 threads.
// ---------------------------------------------------------------------------
__global__ __launch_bounds__(128)
void softmax_kernel(float* __restrict__ attn) {
  __shared__ float sdata[128];
  int row = blockIdx.x;
  float* rp = attn + (size_t)row * CN;
  int tid = threadIdx.x;
  float x = rp[tid];
  sdata[tid] = x;
  __syncthreads();
  for (int s = 64; s > 0; s >>= 1) {
    if (tid < s) sdata[tid] = fmaxf(sdata[tid], sdata[tid + s]);
    __syncthreads();
  }
  float mx = sdata[0];
  __syncthreads();
  float e = __expf(x - mx);
  sdata[tid] = e;
  __syncthreads();
  for (int s = 64; s > 0; s >>= 1) {
    if (tid < s) sdata[tid] += sdata[tid + s];
    __syncthreads();
  }
  rp[tid] = e / sdata[0];
}

__global__ void zero_kernel(float* __restrict__ p, int n) {
  int i = blockIdx.x * 256 + threadIdx.x;
  if (i < n) p[i] = 0.0f;
}

// ---------------------------------------------------------------------------
extern "C" void kernel_launch(void* const* d_in, const int* in_sizes, int n_in,
                              void* d_out, int out_size, void* d_ws, size_t ws_size,
                              hipStream_t stream) {
  (void)in_sizes; (void)n_in; (void)out_size; (void)ws_size;
  const float* u    = (const float*)d_in[0];
  const float* v    = (const float*)d_in[1];
  const float* offw = (const float*)d_in[2];
  const float* offb = (const float*)d_in[3];
  const float* dcnw = (const float*)d_in[4];
  const float* dcnb = (const float*)d_in[5];
  const float* qw   = (const float*)d_in[6];
  const float* qb   = (const float*)d_in[7];
  const float* kw   = (const float*)d_in[8];
  const float* kb   = (const float*)d_in[9];
  const float* vw   = (const float*)d_in[10];
  const float* vb   = (const float*)d_in[11];
  float* out = (float*)d_out;
  float* ws  = (float*)d_ws;

  // workspace layout (floats)
  float* off_all = ws;                                    // 4*18*PN
  float* du   = off_all + (size_t)4 * 18 * PN;            // NBATCH*CN*PN
  float* dv   = du   + (size_t)NBATCH * CN * PN;
  float* qu   = dv   + (size_t)NBATCH * CN * PN;
  float* kv   = qu   + (size_t)NBATCH * CN * PN;
  float* vv   = kv   + (size_t)NBATCH * CN * PN;
  float* attn = vv   + (size_t)NBATCH * CN * PN;          // NBATCH*CN*CN
  float* S    = attn + (size_t)NBATCH * CN * CN;          // RD*PN (reused per image)

  zero_kernel<<<dim3((NBATCH * CN * CN + 255) / 256), 256, 0, stream>>>(attn, NBATCH * CN * CN);
  offset_conv_kernel<<<dim3(PN / 256, 18, 4), 256, 0, stream>>>(u, v, offw, offb, off_all);

  for (int img = 0; img < 4; ++img) {           // t*2+b : u0,u1,v0,v1
    int t = img >> 1, bb = img & 1;
    const float* x = (t ? v : u) + (size_t)bb * CN * PN;
    const float* offp = off_all + (size_t)img * 18 * PN;
    build_S_kernel<<<dim3(PN / 256, KKN, CN), 256, 0, stream>>>(x, offp, S);
    float* dout = (t ? dv : du) + (size_t)bb * CN * PN;
    gemm128_wmma_kernel<<<dim3(PN / 128), 256, 0, stream>>>(dcnw, S, dcnb, dout, RD);
  }

  for (int bb = 0; bb < NBATCH; ++bb) {
    gemm128_wmma_kernel<<<dim3(PN / 128), 256, 0, stream>>>(qw, du + (size_t)bb * CN * PN, qb, qu + (size_t)bb * CN * PN, CN);
    gemm128_wmma_kernel<<<dim3(PN / 128), 256, 0, stream>>>(kw, dv + (size_t)bb * CN * PN, kb, kv + (size_t)bb * CN * PN, CN);
    gemm128_wmma_kernel<<<dim3(PN / 128), 256, 0, stream>>>(vw, dv + (size_t)bb * CN * PN, vb, vv + (size_t)bb * CN * PN, CN);
  }

  attn_logits_kernel<<<dim3(PN / NCHUNK, NBATCH), 256, 0, stream>>>(qu, kv, attn);
  softmax_kernel<<<dim3(NBATCH * CN), 128, 0, stream>>>(attn);

  for (int bb = 0; bb < NBATCH; ++bb) {
    gemm128_wmma_kernel<<<dim3(PN / 128), 256, 0, stream>>>(
        attn + (size_t)bb * CN * CN, vv + (size_t)bb * CN * PN, nullptr,
        out + (size_t)bb * CN * PN, CN);
  }
}